// AttentionHead_3633542332758
// MI455X (gfx1250) — compile-verified
//
#include <hip/hip_runtime.h>
#include <hip/hip_bf16.h>

// DeBERTa disentangled attention head for MI455X (gfx1250), fp32 WMMA path.
// Pipeline (all fp32, memory-bound => no precision downgrade needed):
//   1) proj_kernel:  Q,K,V = x@W*, QR,KR = pos_x@W*r           (WMMA 16x16x4 f32)
//   2) score_gemm:   c2pA = Q@KR^T, p2cT = K@QR^T  (B,S,K2)    (WMMA 16x16x4 f32)
//   3) flash_kernel: c2c via WMMA + diagonal gathers of c2pA/p2cT,
//      online softmax (wave32 16-lane shfl reductions), P@V via WMMA.
// K/V and activation tiles are staged with GLOBAL_LOAD_ASYNC_TO_LDS_B128 and
// double-buffered (ASYNCcnt / s_wait_asynccnt) when the toolchain exposes the
// builtins; otherwise falls back to plain float4 LDS stores.
// Workspace requirement: 5*B*S*DH + 2*B*S*K2 floats = ~276 MB in d_ws.

typedef __attribute__((ext_vector_type(2))) float v2f;
typedef __attribute__((ext_vector_type(8))) float v8f;
typedef __attribute__((ext_vector_type(4))) int   v4i;

#define BB 8
#define SS 2048
#define DM 1024
#define DH 64
#define KK2 2048

#if defined(__has_builtin)
#if __has_builtin(__builtin_amdgcn_global_load_async_to_lds_b128) && \
    __has_builtin(__builtin_amdgcn_s_wait_asynccnt)
#define ASYNC_LDS 1
#endif
#endif

#ifdef ASYNC_LDS
typedef __attribute__((address_space(1))) v4i as1_v4i;  // global int32x4
typedef __attribute__((address_space(3))) v4i as3_v4i;  // LDS int32x4
static __device__ __forceinline__ void async_ld_b128(const void* g, void* l) {
  // global -> LDS, 16 bytes per lane, tracked with ASYNCcnt
  __builtin_amdgcn_global_load_async_to_lds_b128((as1_v4i*)(void*)g,
                                                 (as3_v4i*)l, 0, 0);
}
static __device__ __forceinline__ void wait_async0() {
  __builtin_amdgcn_s_wait_asynccnt(0);
}
#else
static __device__ __forceinline__ void wait_async0() {}
#endif

// Copy `n4` float4 elements from contiguous global to contiguous LDS using all
// `nthr` threads of the block. 16B aligned on both sides.
static __device__ __forceinline__ void stage_lin(const float* __restrict__ g,
                                                 float* __restrict__ l,
                                                 int n4, int tid, int nthr) {
#ifdef ASYNC_LDS
  for (int e = tid; e < n4; e += nthr) async_ld_b128(g + e * 4, l + e * 4);
#else
  for (int e = tid; e < n4; e += nthr)
    ((float4*)l)[e] = ((const float4*)g)[e];
#endif
}

static __device__ __forceinline__ v8f wmma_f32(v2f a, v2f b, v8f c) {
  // V_WMMA_F32_16X16X4_F32 : D = A(16x4) * B(4x16) + C(16x16), fp32 throughout
  return __builtin_amdgcn_wmma_f32_16x16x4_f32(
      /*neg_a=*/false, a, /*neg_b=*/false, b,
      /*c_mod=*/(short)0, c, /*reuse_a=*/false, /*reuse_b=*/false);
}

// ---------------------------------------------------------------------------
// Kernel 1: projections. grid = (M/16, 5), block = 128 (4 waves, N=64 cols).
// which: 0=Q,1=K,2=V from x; 3=QR,4=KR from pos_x. M = B*S = B*K2 = 16384.
// Activation chunks (16 rows x 128 K) double-buffered through LDS.
// ---------------------------------------------------------------------------
__global__ __launch_bounds__(128) void proj_kernel(
    const float* __restrict__ x, const float* __restrict__ pos_x,
    const float* __restrict__ Wq, const float* __restrict__ bq,
    const float* __restrict__ Wk, const float* __restrict__ bk,
    const float* __restrict__ Wv, const float* __restrict__ bv,
    const float* __restrict__ Wqr, const float* __restrict__ bqr,
    const float* __restrict__ Wkr, const float* __restrict__ bkr,
    float* __restrict__ Qb, float* __restrict__ Kb, float* __restrict__ Vb,
    float* __restrict__ QRb, float* __restrict__ KRb) {
  __shared__ float Xl[2][16 * 128];  // 2 x 8KB double-buffered activation tile

  const int which = blockIdx.y;
  const float* W;
  const float* bias;
  const float* src;
  float* dst;
  switch (which) {
    case 0: W = Wq;  bias = bq;  src = x;     dst = Qb;  break;
    case 1: W = Wk;  bias = bk;  src = x;     dst = Kb;  break;
    case 2: W = Wv;  bias = bv;  src = x;     dst = Vb;  break;
    case 3: W = Wqr; bias = bqr; src = pos_x; dst = QRb; break;
    default: W = Wkr; bias = bkr; src = pos_x; dst = KRb; break;
  }

  const int m0 = blockIdx.x * 16;
  const int tid = threadIdx.x;
  const int lane = tid & 31, wave = tid >> 5;
  const int half = lane >> 4, l16 = lane & 15;
  const int ncol = wave * 16 + l16;  // this wave's output column

  // stage: rows are NOT contiguous (row stride DM), copy 128-float rows
  auto stage_chunk = [&](int kc, float* dstl) {
    for (int e = tid; e < 16 * 32; e += 128) {  // 512 float4s
      const int row = e >> 5, c4 = e & 31;
#ifdef ASYNC_LDS
      async_ld_b128(src + (size_t)(m0 + row) * DM + kc + c4 * 4,
                    dstl + row * 128 + c4 * 4);
#else
      ((float4*)(dstl + row * 128))[c4] =
          ((const float4*)(src + (size_t)(m0 + row) * DM + kc))[c4];
#endif
    }
  };

  stage_chunk(0, Xl[0]);  // prologue

  v8f acc = {};
  for (int c = 0; c < DM / 128; ++c) {
    const int cur = c & 1;
    wait_async0();
    __syncthreads();
    if (c + 1 < DM / 128) stage_chunk((c + 1) * 128, Xl[cur ^ 1]);

    const float* Xc = Xl[cur];
    const int kc = c * 128;
    for (int s = 0; s < 32; ++s) {
      const int k0 = 4 * s + half * 2;  // A layout: lanes 16-31 take K=+2
      v2f a;
      a.x = Xc[l16 * 128 + k0];
      a.y = Xc[l16 * 128 + k0 + 1];
      v2f b;  // B layout: lane = N column, half selects K pair
      b.x = W[(size_t)(kc + k0) * DH + ncol];
      b.y = W[(size_t)(kc + k0 + 1) * DH + ncol];
      acc = wmma_f32(a, b, acc);
    }
  }
  const float bsv = bias[ncol];
  for (int r = 0; r < 8; ++r) {  // C layout: reg r -> row r (+8 for upper half)
    const int m = m0 + r + half * 8;
    dst[(size_t)m * DH + ncol] = acc[r] + bsv;
  }
}

// ---------------------------------------------------------------------------
// Kernel 2: (2048x64)@(2048x64)^T -> (2048x2048) per (batch, which).
// which 0: c2pA = Q @ KR^T ; which 1: p2cT = K @ QR^T (transposed p2c layout
// so both gathers in kernel 3 become out[row, pos]).
// grid = (32*32, 2, B), block = 128 (4 waves -> 64x64 tile).
// ---------------------------------------------------------------------------
__global__ __launch_bounds__(128) void score_gemm(
    const float* __restrict__ Qb, const float* __restrict__ Kb,
    const float* __restrict__ QRb, const float* __restrict__ KRb,
    float* __restrict__ c2pA, float* __restrict__ p2cT) {
  __shared__ float Al[64 * 64];  // 16KB
  __shared__ float Bl[64 * 64];  // 16KB

  const int b = blockIdx.z;
  const int which = blockIdx.y;
  const float* A = (which ? Kb : Qb) + (size_t)b * SS * DH;
  const float* Bm = (which ? QRb : KRb) + (size_t)b * KK2 * DH;
  float* outp = (which ? p2cT : c2pA) + (size_t)b * SS * KK2;

  const int it = blockIdx.x >> 5, jt = blockIdx.x & 31;
  const int i0 = it * 64, j0 = jt * 64;
  const int tid = threadIdx.x;
  const int lane = tid & 31, wave = tid >> 5;
  const int half = lane >> 4, l16 = lane & 15;

  // 64 rows x 64 cols each, contiguous in memory (row stride == DH)
  stage_lin(A + (size_t)i0 * DH, Al, 64 * 16, tid, 128);
  stage_lin(Bm + (size_t)j0 * DH, Bl, 64 * 16, tid, 128);
  wait_async0();
  __syncthreads();

  const int mrow = wave * 16 + l16;  // A row for this lane (within 64-tile)
  for (int t = 0; t < 4; ++t) {      // 4 N-subtiles of 16
    v8f acc = {};
    for (int s = 0; s < 16; ++s) {   // K = 64 = 16 steps of 4
      const int k0 = 4 * s + half * 2;
      v2f a;
      a.x = Al[mrow * 64 + k0];
      a.y = Al[mrow * 64 + k0 + 1];
      v2f bv;  // B[k][n] = Bm[j0+n][k]  (B^T GEMM)
      bv.x = Bl[(t * 16 + l16) * 64 + k0];
      bv.y = Bl[(t * 16 + l16) * 64 + k0 + 1];
      acc = wmma_f32(a, bv, acc);
    }
    for (int r = 0; r < 8; ++r) {
      const int mi = i0 + wave * 16 + r + half * 8;
      const int nj = j0 + t * 16 + l16;
      outp[(size_t)mi * KK2 + nj] = acc[r];
    }
  }
}

// ---------------------------------------------------------------------------
// Kernel 3: flash attention with disentangled-bias gathers.
// grid = (S/64, B), block = 128: 4 waves, each owns 16 query rows.
// K/V tiles double-buffered via async-to-LDS; one barrier per j-iteration.
// ---------------------------------------------------------------------------
__global__ __launch_bounds__(128) void flash_kernel(
    const float* __restrict__ Qb, const float* __restrict__ Kb,
    const float* __restrict__ Vb, const float* __restrict__ c2pA,
    const float* __restrict__ p2cT, const int* __restrict__ mask,
    float* __restrict__ out) {
  __shared__ float Kl[2][16 * 64];    // 2 x 4KB key tiles
  __shared__ float Vl[2][16 * 64];    // 2 x 4KB value tiles
  __shared__ float Pl[4 * 16 * 16];   // 4KB  per-wave P re-layout scratch

  const int b = blockIdx.y;
  const int tid = threadIdx.x;
  const int lane = tid & 31, wave = tid >> 5;
  const int half = lane >> 4, l16 = lane & 15;

  const float* Q = Qb + (size_t)b * SS * DH;
  const float* K = Kb + (size_t)b * SS * DH;
  const float* V = Vb + (size_t)b * SS * DH;
  const float* cA = c2pA + (size_t)b * SS * KK2;
  const float* pT = p2cT + (size_t)b * SS * KK2;
  const int* mk = mask + (size_t)b * SS;

  const int iBase = blockIdx.x * 64 + wave * 16;

  // Q rows for this wave, pre-packed in WMMA A-layout registers (16 k-steps).
  v2f qa[16];
  {
    const int i = iBase + l16;
    for (int s = 0; s < 16; ++s) {
      const int k0 = 4 * s + half * 2;
      qa[s].x = Q[(size_t)i * DH + k0];
      qa[s].y = Q[(size_t)i * DH + k0 + 1];
    }
  }

  v8f o0 = {}, o1 = {}, o2 = {}, o3 = {};
  float mrun[8], lrun[8];
  for (int r = 0; r < 8; ++r) { mrun[r] = -1e30f; lrun[r] = 0.0f; }
  const float scale = 0.07216878364870323f;  // 1/sqrt(3*64)

  float* Pw = Pl + wave * 256;

  // prologue: stage j-block 0 into buffer 0 (16 rows x 64 = contiguous 1024 f)
  stage_lin(K, Kl[0], 256, tid, 128);
  stage_lin(V, Vl[0], 256, tid, 128);

  for (int jb = 0; jb < SS / 16; ++jb) {
    const int cur = jb & 1;
    wait_async0();
    __syncthreads();  // buffer `cur` ready for everyone
    if (jb + 1 < SS / 16) {
      // buffer cur^1 was last read before the barrier above -> safe to refill
      const size_t j1 = (size_t)(jb + 1) * 16 * DH;
      stage_lin(K + j1, Kl[cur ^ 1], 256, tid, 128);
      stage_lin(V + j1, Vl[cur ^ 1], 256, tid, 128);
    }
    const float* Kc = Kl[cur];
    const float* Vc = Vl[cur];
    const int j0 = jb * 16;

    // c2c tile (16x16) = Q_wave @ K_tile^T
    v8f sc = {};
    for (int s = 0; s < 16; ++s) {
      const int k0 = 4 * s + half * 2;
      v2f bv;
      bv.x = Kc[l16 * 64 + k0];
      bv.y = Kc[l16 * 64 + k0 + 1];
      sc = wmma_f32(qa[s], bv, sc);
    }

    const int j = j0 + l16;          // this lane's key column
    const int msk = mk[j];
    for (int r = 0; r < 8; ++r) {
      const int i = iBase + r + half * 8;
      int p = i - j + KK2 / 2;
      p = p < 0 ? 0 : (p > KK2 - 1 ? KK2 - 1 : p);
      float sv = (sc[r] + cA[(size_t)i * KK2 + p] + pT[(size_t)j * KK2 + p]) * scale;
      if (msk == 1) sv = -1e30f;

      // row reductions: a row spans the 16 lanes of this half (xor<=8 stays in half)
      float tm = sv;
      tm = fmaxf(tm, __shfl_xor(tm, 1, 32));
      tm = fmaxf(tm, __shfl_xor(tm, 2, 32));
      tm = fmaxf(tm, __shfl_xor(tm, 4, 32));
      tm = fmaxf(tm, __shfl_xor(tm, 8, 32));
      const float mnew = fmaxf(mrun[r], tm);
      const float sold = __expf(mrun[r] - mnew);
      const float pv = __expf(sv - mnew);
      float rs = pv;
      rs += __shfl_xor(rs, 1, 32);
      rs += __shfl_xor(rs, 2, 32);
      rs += __shfl_xor(rs, 4, 32);
      rs += __shfl_xor(rs, 8, 32);
      lrun[r] = lrun[r] * sold + rs;
      mrun[r] = mnew;
      o0[r] *= sold; o1[r] *= sold; o2[r] *= sold; o3[r] *= sold;
      Pw[(r + half * 8) * 16 + l16] = pv;  // row-major P tile in LDS
    }

    // O += P(16x16) @ V_tile(16x64): re-read P in A-layout from LDS.
    for (int s = 0; s < 4; ++s) {
      const int k0 = 4 * s + half * 2;
      v2f a;
      a.x = Pw[l16 * 16 + k0];
      a.y = Pw[l16 * 16 + k0 + 1];
      v2f b0, b1, b2, b3;
      b0.x = Vc[k0 * 64 + l16];       b0.y = Vc[(k0 + 1) * 64 + l16];
      b1.x = Vc[k0 * 64 + 16 + l16];  b1.y = Vc[(k0 + 1) * 64 + 16 + l16];
      b2.x = Vc[k0 * 64 + 32 + l16];  b2.y = Vc[(k0 + 1) * 64 + 32 + l16];
      b3.x = Vc[k0 * 64 + 48 + l16];  b3.y = Vc[(k0 + 1) * 64 + 48 + l16];
      o0 = wmma_f32(a, b0, o0);
      o1 = wmma_f32(a, b1, o1);
      o2 = wmma_f32(a, b2, o2);
      o3 = wmma_f32(a, b3, o3);
    }
  }

  float* op = out + (size_t)b * SS * DH;
  for (int r = 0; r < 8; ++r) {
    const int i = iBase + r + half * 8;
    const float inv = 1.0f / lrun[r];
    op[(size_t)i * DH + l16]      = o0[r] * inv;
    op[(size_t)i * DH + 16 + l16] = o1[r] * inv;
    op[(size_t)i * DH + 32 + l16] = o2[r] * inv;
    op[(size_t)i * DH + 48 + l16] = o3[r] * inv;
  }
}

// ---------------------------------------------------------------------------
extern "C" void kernel_launch(void* const* d_in, const int* in_sizes, int n_in,
                              void* d_out, int out_size, void* d_ws, size_t ws_size,
                              hipStream_t stream) {
  (void)in_sizes; (void)n_in; (void)out_size; (void)ws_size;
  const float* x     = (const float*)d_in[0];
  const float* pos_x = (const float*)d_in[1];
  const int*   pmask = (const int*)d_in[2];
  const float* Wq  = (const float*)d_in[3];
  const float* bq  = (const float*)d_in[4];
  const float* Wk  = (const float*)d_in[5];
  const float* bk  = (const float*)d_in[6];
  const float* Wv  = (const float*)d_in[7];
  const float* bv  = (const float*)d_in[8];
  const float* Wqr = (const float*)d_in[9];
  const float* bqr = (const float*)d_in[10];
  const float* Wkr = (const float*)d_in[11];
  const float* bkr = (const float*)d_in[12];

  float* ws = (float*)d_ws;
  const size_t PROJ = (size_t)BB * SS * DH;     // 1,048,576 floats each
  const size_t ATT  = (size_t)BB * SS * KK2;    // 33,554,432 floats each
  float* Qb   = ws;
  float* Kb   = Qb + PROJ;
  float* Vb   = Kb + PROJ;
  float* QRb  = Vb + PROJ;
  float* KRb  = QRb + PROJ;
  float* c2pA = KRb + PROJ;
  float* p2cT = c2pA + ATT;   // total ws need: 5*PROJ + 2*ATT floats (~276 MB)

  proj_kernel<<<dim3((BB * SS) / 16, 5), 128, 0, stream>>>(
      x, pos_x, Wq, bq, Wk, bk, Wv, bv, Wqr, bqr, Wkr, bkr,
      Qb, Kb, Vb, QRb, KRb);

  score_gemm<<<dim3((SS / 64) * (KK2 / 64), 2, BB), 128, 0, stream>>>(
      Qb, Kb, QRb, KRb, c2pA, p2cT);

  flash_kernel<<<dim3(SS / 64, BB), 128, 0, stream>>>(
      Qb, Kb, Vb, c2pA, p2cT, pmask, (float*)d_out);
}